// GenderAwareCrossEntropyLoss_15333033246942
// MI455X (gfx1250) — compile-verified
//
#include <hip/hip_runtime.h>

// GenderAwareCrossEntropyLoss for MI455X (gfx1250, wave32).
// Memory-bound streaming reduction: 160 MB in, 4 B out. No GEMM -> no WMMA.
// Strategy: 4 rows (= 112 B = 7 x float4, 16B aligned) per thread so ALL
// traffic is global_load_b128; gfx1250 global_prefetch_b8 on the next tile;
// wave32 shuffle + LDS tree reduction; deterministic two-pass mean.

#define NCLS 7
#define GENDER_PENALTY 5.0f
// VALID_TABLE packed: idx = g1*2+g2 -> 7-bit class mask, 8 bits per idx.
// (0,0)->{1,4}=0x12  (0,1)->{0,3,6}=0x49  (1,0)->{0,3,6}=0x49  (1,1)->{2,5}=0x24
#define VALID_PACKED 0x24494912u

__device__ __forceinline__ float wave32_sum(float v) {
#pragma unroll
    for (int off = 16; off > 0; off >>= 1)
        v += __shfl_down(v, off, 32);   // wave32: width must be 32 on gfx1250
    return v;
}

// Reduce a per-thread value across the block; lane0/wave0 gets the total.
template <int MAXWAVES>
__device__ __forceinline__ float block_sum(float v) {
    __shared__ float sm[MAXWAVES];
    const int lane = threadIdx.x & 31;
    const int wid  = threadIdx.x >> 5;
    float ws = wave32_sum(v);
    if (lane == 0) sm[wid] = ws;
    __syncthreads();
    float out = 0.0f;
    if (wid == 0) {
        float x = (lane < (int)(blockDim.x >> 5)) ? sm[lane] : 0.0f;
        out = wave32_sum(x);
    }
    return out;  // valid in (wid==0, lane==0)
}

__device__ __forceinline__ float row_loss(const float x[NCLS], int lab,
                                          int g1, int g2,
                                          float w0, float w1, float w2, float w3,
                                          float w4, float w5, float w6) {
    // max + argmax (first-max tie break, matches jnp.argmax) + label gather,
    // all via unrolled compares -> v_cndmask (no scratch arrays).
    float m = x[0];
    int   am = 0;
    float xl = x[0];
#pragma unroll
    for (int j = 1; j < NCLS; ++j) {
        if (x[j] > m) { m = x[j]; am = j; }
        if (j == lab) xl = x[j];
    }
    float s = 0.0f;
#pragma unroll
    for (int j = 0; j < NCLS; ++j)
        s += __expf(x[j] - m);          // v_exp_f32 (TRANS unit)
    const float logZ = m + __logf(s);   // v_log_f32
    const float ce   = logZ - xl;

    float wt = w0;                       // class_weights[lab] via cndmask chain
    if (lab == 1) wt = w1;
    if (lab == 2) wt = w2;
    if (lab == 3) wt = w3;
    if (lab == 4) wt = w4;
    if (lab == 5) wt = w5;
    if (lab == 6) wt = w6;

    const unsigned mask = (VALID_PACKED >> (((g1 << 1) | g2) << 3)) & 0x7Fu;
    const float pen = ((mask >> am) & 1u) ? 0.0f : GENDER_PENALTY;
    return wt * ce + pen;
}

__global__ void __launch_bounds__(256)
gacel_partials_kernel(const float* __restrict__ logits,
                      const float* __restrict__ cw,
                      const int*   __restrict__ labels,
                      const int*   __restrict__ gender,
                      float*       __restrict__ partials,
                      int n_rows) {
    const int ngroups = n_rows >> 2;              // 4 rows per group
    const int tail0   = ngroups << 2;
    const int stride  = gridDim.x * blockDim.x;
    const int tid     = blockIdx.x * blockDim.x + threadIdx.x;

    // class_weights: uniform address + constant offsets -> scalar loads.
    const float w0 = cw[0], w1 = cw[1], w2 = cw[2], w3 = cw[3];
    const float w4 = cw[4], w5 = cw[5], w6 = cw[6];

    const float4* __restrict__ lp4 = (const float4*)logits;   // 7 float4 / group
    const int4*   __restrict__ lb4 = (const int4*)labels;     // 1 int4   / group
    const int4*   __restrict__ gf4 = (const int4*)gender;     // 2 int4   / group

    float acc = 0.0f;

    for (int g = tid; g < ngroups; g += stride) {
        // gfx1250 prefetch of the next grid-stride tile (global_prefetch_b8).
        if (g + stride < ngroups)
            __builtin_prefetch((const void*)(lp4 + (size_t)(g + stride) * 7), 0, 0);

        float4 q0 = lp4[(size_t)g * 7 + 0];
        float4 q1 = lp4[(size_t)g * 7 + 1];
        float4 q2 = lp4[(size_t)g * 7 + 2];
        float4 q3 = lp4[(size_t)g * 7 + 3];
        float4 q4 = lp4[(size_t)g * 7 + 4];
        float4 q5 = lp4[(size_t)g * 7 + 5];
        float4 q6 = lp4[(size_t)g * 7 + 6];

        const int4 lb  = lb4[g];
        const int4 gfa = gf4[(size_t)g * 2 + 0];   // rows 0,1: (g1,g2,g1,g2)
        const int4 gfb = gf4[(size_t)g * 2 + 1];   // rows 2,3

        {   // row 0: elems 0..6
            const float x[NCLS] = {q0.x, q0.y, q0.z, q0.w, q1.x, q1.y, q1.z};
            acc += row_loss(x, lb.x, gfa.x, gfa.y, w0, w1, w2, w3, w4, w5, w6);
        }
        {   // row 1: elems 7..13
            const float x[NCLS] = {q1.w, q2.x, q2.y, q2.z, q2.w, q3.x, q3.y};
            acc += row_loss(x, lb.y, gfa.z, gfa.w, w0, w1, w2, w3, w4, w5, w6);
        }
        {   // row 2: elems 14..20
            const float x[NCLS] = {q3.z, q3.w, q4.x, q4.y, q4.z, q4.w, q5.x};
            acc += row_loss(x, lb.z, gfb.x, gfb.y, w0, w1, w2, w3, w4, w5, w6);
        }
        {   // row 3: elems 21..27
            const float x[NCLS] = {q5.y, q5.z, q5.w, q6.x, q6.y, q6.z, q6.w};
            acc += row_loss(x, lb.w, gfb.z, gfb.w, w0, w1, w2, w3, w4, w5, w6);
        }
    }

    // Tail rows (n_rows % 4 != 0); scalar path, first few threads only.
    const int tail = n_rows - tail0;
    if (tid < tail) {
        const int row = tail0 + tid;
        float x[NCLS];
#pragma unroll
        for (int j = 0; j < NCLS; ++j) x[j] = logits[(size_t)row * NCLS + j];
        acc += row_loss(x, labels[row], gender[(size_t)row * 2 + 0],
                        gender[(size_t)row * 2 + 1],
                        w0, w1, w2, w3, w4, w5, w6);
    }

    const float bs = block_sum<8>(acc);            // 256 threads = 8 waves
    if (threadIdx.x == 0) partials[blockIdx.x] = bs;
}

__global__ void __launch_bounds__(1024)
gacel_finalize_kernel(const float* __restrict__ partials, int nparts,
                      float* __restrict__ out, float inv_n) {
    float acc = 0.0f;
    for (int i = threadIdx.x; i < nparts; i += blockDim.x)
        acc += partials[i];
    const float total = block_sum<32>(acc);        // 1024 threads = 32 waves
    if (threadIdx.x == 0) out[0] = total * inv_n;
}

extern "C" void kernel_launch(void* const* d_in, const int* in_sizes, int n_in,
                              void* d_out, int out_size, void* d_ws, size_t ws_size,
                              hipStream_t stream) {
    const float* logits = (const float*)d_in[0];   // [N, 7] f32
    const float* cw     = (const float*)d_in[1];   // [7]    f32
    const int*   labels = (const int*)d_in[2];     // [N]    i32
    const int*   gender = (const int*)d_in[3];     // [N, 2] i32
    float*       out    = (float*)d_out;
    float*       parts  = (float*)d_ws;

    const int n_rows  = in_sizes[2];
    const int ngroups = n_rows / 4;

    const int threads = 256;
    int blocks = (ngroups + threads - 1) / threads;
    if (blocks < 1) blocks = 1;
    if (blocks > 2048) blocks = 2048;              // fixed partition -> deterministic

    gacel_partials_kernel<<<blocks, threads, 0, stream>>>(
        logits, cw, labels, gender, parts, n_rows);

    gacel_finalize_kernel<<<1, 1024, 0, stream>>>(
        parts, blocks, out, 1.0f / (float)n_rows);
}